// NoisyPQC_22402549416579
// MI455X (gfx1250) — compile-verified
//
#include <hip/hip_runtime.h>
#include <hip/hip_bf16.h>

#define N_QUBITS 12
#define DIM      4096
#define N_LAYERS 4
#define REPS     200
#define BATCH    16
#define THREADS  256   // 8 wave32s

typedef __attribute__((ext_vector_type(2))) float v2f;
typedef __attribute__((ext_vector_type(8))) float v8f;

// One workgroup simulates one (batch, rep) trajectory with the 4096-amplitude
// state held in LDS as split re/im arrays. Readout <Z_q> is done with chained
// V_WMMA_F32_16X16X4_F32 over probs reshaped as a 256x16 matrix.
__global__ __launch_bounds__(THREADS)
void noisy_pqc_traj(const float* __restrict__ angles,   // [16,12]
                    const float* __restrict__ params,   // [4,12,2]
                    const int*   __restrict__ noise,    // [200,4,12]
                    float*       __restrict__ out)      // [16,12]
{
    __shared__ float s_re[DIM];
    __shared__ float s_im[DIM];
    __shared__ float s_gates[N_LAYERS * N_QUBITS * 8]; // fused Rx*Rz, complex 2x2
    __shared__ float s_cs[2 * N_QUBITS];               // cos(a/2), sin(a/2)
    __shared__ float s_red[8 * 8 * 32];                // per-wave WMMA D tiles

    const int tid  = threadIdx.x;
    const int b    = blockIdx.x & (BATCH - 1);
    const int rep  = blockIdx.x >> 4;

    // ---- build gate table: G = Rx(t1) * Rz(t0), one gate per thread < 48 ----
    if (tid < N_LAYERS * N_QUBITS) {
        const int l = tid / N_QUBITS, q = tid % N_QUBITS;
        const float t0 = params[(l * N_QUBITS + q) * 2 + 0];
        const float t1 = params[(l * N_QUBITS + q) * 2 + 1];
        const float x0 = cosf(0.5f * t0), y0 = -sinf(0.5f * t0); // e0 = exp(-i t0/2)
        const float c  = cosf(0.5f * t1), s  = sinf(0.5f * t1);
        float* g = &s_gates[tid * 8];
        // G00 = c*e0 ; G01 = -i*s*e1 ; G10 = -i*s*e0 ; G11 = c*e1  (e1 = conj(e0))
        g[0] = c * x0;   g[1] = c * y0;      // G00
        g[2] = -s * y0;  g[3] = -s * x0;     // G01 = s*(e1.im, -e1.re) = (-s*y0, -s*x0)
        g[4] = s * y0;   g[5] = -s * x0;     // G10 = s*(e0.im, -e0.re)
        g[6] = c * x0;   g[7] = -c * y0;     // G11
    }
    if (tid >= 48 && tid < 48 + N_QUBITS) {
        const int q = tid - 48;
        const float a = angles[b * N_QUBITS + q];
        s_cs[2 * q]     = cosf(0.5f * a);
        s_cs[2 * q + 1] = sinf(0.5f * a);
    }
    __syncthreads();

    // ---- initial product state: amp[idx] = (prod c/s) * (-i)^popcount(idx) ----
    for (int j = 0; j < DIM / THREADS; ++j) {
        const int idx = tid + THREADS * j;
        float r = 1.0f;
        #pragma unroll
        for (int q = 0; q < N_QUBITS; ++q)
            r *= ((idx >> (11 - q)) & 1) ? s_cs[2 * q + 1] : s_cs[2 * q];
        const int pc = __popc(idx) & 3;
        s_re[idx] = (pc == 0) ? r : ((pc == 2) ? -r : 0.0f);
        s_im[idx] = (pc == 3) ? r : ((pc == 1) ? -r : 0.0f);
    }
    __syncthreads();

    // ---- layers: fused (Rx*Rz*PrevPauli) sweeps; CZ diag fused into q==11 ----
    for (int l = 0; l < N_LAYERS; ++l) {
        for (int q = 0; q < N_QUBITS; ++q) {
            const float* g = &s_gates[(l * N_QUBITS + q) * 8];
            float m00r = g[0], m00i = g[1], m01r = g[2], m01i = g[3];
            float m10r = g[4], m10i = g[5], m11r = g[6], m11i = g[7];
            if (l > 0) {
                const int p = noise[((rep * N_LAYERS + (l - 1)) * N_QUBITS) + q];
                if (p == 1) {                       // X: swap columns
                    float t;
                    t = m00r; m00r = m01r; m01r = t;  t = m00i; m00i = m01i; m01i = t;
                    t = m10r; m10r = m11r; m11r = t;  t = m10i; m10i = m11i; m11i = t;
                } else if (p == 2) {                // Y: col0 = i*G01, col1 = -i*G00
                    float a0r = m00r, a0i = m00i, a1r = m01r, a1i = m01i;
                    float b0r = m10r, b0i = m10i, b1r = m11r, b1i = m11i;
                    m00r = -a1i; m00i =  a1r;   m01r =  a0i; m01i = -a0r;
                    m10r = -b1i; m10i =  b1r;   m11r =  b0i; m11i = -b0r;
                } else if (p == 3) {                // Z: negate column 1
                    m01r = -m01r; m01i = -m01i; m11r = -m11r; m11i = -m11i;
                }
            }
            const int bq = 11 - q;
            const int lowmask = (1 << bq) - 1;
            #pragma unroll 2
            for (int j = 0; j < (DIM / 2) / THREADS; ++j) {
                const int p2 = tid + THREADS * j;
                const int i0 = ((p2 >> bq) << (bq + 1)) | (p2 & lowmask);
                const int i1 = i0 | (1 << bq);
                const float ar = s_re[i0], ai = s_im[i0];
                const float br = s_re[i1], bi = s_im[i1];
                float nar = m00r * ar - m00i * ai + m01r * br - m01i * bi;
                float nai = m00r * ai + m00i * ar + m01r * bi + m01i * br;
                float nbr = m10r * ar - m10i * ai + m11r * br - m11i * bi;
                float nbi = m10r * ai + m10i * ar + m11r * bi + m11i * br;
                if (q == 11) { // CZ chain: (-1)^popcount(idx & idx>>1)
                    const float cz0 = 1.0f - 2.0f * (float)(__popc(i0 & (i0 >> 1)) & 1);
                    const float cz1 = 1.0f - 2.0f * (float)(__popc(i1 & (i1 >> 1)) & 1);
                    nar *= cz0; nai *= cz0; nbr *= cz1; nbi *= cz1;
                }
                s_re[i0] = nar; s_im[i0] = nai;
                s_re[i1] = nbr; s_im[i1] = nbi;
            }
            __syncthreads();
        }
    }

    // ---- WMMA readout: D = W(16xK) * P(Kx16), P[u][v] = probs[16u+v] --------
    // Row 0 of W = 1 (column sums -> qubits 8..11), rows 1..8 = Z-sign of
    // qubit (m-1) over u (row sums of D -> qubits 0..7). K tiled by 4, wave w
    // covers u in [32w, 32w+32).
    const int lane  = tid & 31;
    const int w     = tid >> 5;
    const int vcol  = lane & 15;
    const int k0    = (lane < 16) ? 0 : 2;
    const int m     = lane & 15;
    const bool is0  = (m == 0);
    const bool mid  = (m >= 1 && m <= 8);
    const int  shv  = mid ? (8 - m) : 0;

    v8f acc;
    #pragma unroll
    for (int i = 0; i < 8; ++i) acc[i] = 0.0f;

    for (int t = 0; t < 8; ++t) {
        const int u0 = w * 32 + t * 4 + k0;
        const int u1 = u0 + 1;
        v2f A, B;
        A.x = is0 ? 1.0f : (mid ? (1.0f - 2.0f * (float)((u0 >> shv) & 1)) : 0.0f);
        A.y = is0 ? 1.0f : (mid ? (1.0f - 2.0f * (float)((u1 >> shv) & 1)) : 0.0f);
        const int ia = u0 * 16 + vcol, ib = u1 * 16 + vcol;
        B.x = s_re[ia] * s_re[ia] + s_im[ia] * s_im[ia];
        B.y = s_re[ib] * s_re[ib] + s_im[ib] * s_im[ib];
        acc = __builtin_amdgcn_wmma_f32_16x16x4_f32(
                  false, A, false, B, (short)0, acc, false, false);
    }
    #pragma unroll
    for (int r = 0; r < 8; ++r)
        s_red[w * 256 + r * 32 + lane] = acc[r];
    __syncthreads();

    // ---- final 12 sums + layer-3 Pauli sign flip + mean accumulation --------
    if (tid < N_QUBITS) {
        const int q = tid;
        float S = 0.0f;
        if (q < 8) {
            const int mm  = q + 1;
            const int rr  = mm & 7;
            const int off = (mm >= 8) ? 16 : 0;
            for (int ww = 0; ww < 8; ++ww)
                for (int v = 0; v < 16; ++v)
                    S += s_red[ww * 256 + rr * 32 + off + v];
        } else {
            for (int ww = 0; ww < 8; ++ww)
                for (int v = 0; v < 16; ++v) {
                    const float z = 1.0f - 2.0f * (float)((v >> (11 - q)) & 1);
                    S += z * s_red[ww * 256 + v];
                }
        }
        const int p3 = noise[((rep * N_LAYERS + 3) * N_QUBITS) + q];
        if (p3 == 1 || p3 == 2) S = -S;   // X/Y in last layer flips <Z_q>
        atomicAdd(&out[b * N_QUBITS + q], S * (1.0f / (float)REPS));
    }
}

__global__ void zero_out_kernel(float* out, int n) {
    const int i = blockIdx.x * blockDim.x + threadIdx.x;
    if (i < n) out[i] = 0.0f;
}

extern "C" void kernel_launch(void* const* d_in, const int* in_sizes, int n_in,
                              void* d_out, int out_size, void* d_ws, size_t ws_size,
                              hipStream_t stream) {
    const float* angles = (const float*)d_in[0];   // [16,12]
    const float* params = (const float*)d_in[1];   // [4,12,2]
    const int*   noise  = (const int*)d_in[2];     // [200,4,12]
    float* out = (float*)d_out;                    // [16,12]

    zero_out_kernel<<<1, 256, 0, stream>>>(out, out_size);
    noisy_pqc_traj<<<BATCH * REPS, THREADS, 0, stream>>>(angles, params, noise, out);
}